// RadialSpectralConv_13718125543613
// MI455X (gfx1250) — compile-verified
//
#include <hip/hip_runtime.h>
#include <hip/hip_bf16.h>

typedef __attribute__((ext_vector_type(2))) float v2f;
typedef __attribute__((ext_vector_type(8))) float v8f;

#define PI_F 3.14159265358979323846f

// signed frequency for kept-mode row r (r in [0,64)): idx = r<32 ? r : r+64 ; freq = r<32 ? r : r-64
__device__ __forceinline__ int sfreq(int r) { return r < 32 ? r : r - 64; }

// ---- WMMA f32 16x16x4 helpers (CDNA5 VGPR layouts, ISA 7.12.2) -------------
// A is 16x4 fp32: lane l holds M=l%16, K pair base = 2*(l/16): a.x=K+0, a.y=K+1
__device__ __forceinline__ v2f load_a(const float* A, int ld, int k0, int lane) {
  int m = lane & 15;
  int kk = k0 + ((lane >> 4) << 1);
  const float* p = A + m * ld + kk;
  v2f a; a.x = p[0]; a.y = p[1];
  return a;
}
// B is 4x16 fp32: lane l holds N=l%16; b.x = row k0+(l/16), b.y = row k0+2+(l/16)
__device__ __forceinline__ v2f load_b(const float* B, int ld, int k0, int n0, int lane) {
  int n = n0 + (lane & 15);
  int kr = k0 + (lane >> 4);
  v2f b;
  b.x = B[kr * ld + n];
  b.y = B[(kr + 2) * ld + n];
  return b;
}
__device__ __forceinline__ v8f wmma4(v2f a, v2f b, v8f c) {
  return __builtin_amdgcn_wmma_f32_16x16x4_f32(false, a, false, b, (short)0, c, false, false);
}

// ---- twiddle init: Wf (64x128, fwd, 1/128 norm) and Wi (128x64, inverse) ---
__global__ __launch_bounds__(256) void init_twiddles(float* __restrict__ wf_re, float* __restrict__ wf_im,
                                                     float* __restrict__ wi_re, float* __restrict__ wi_im) {
  int t = blockIdx.x * blockDim.x + threadIdx.x;
  if (t < 64 * 128) {
    int r = t >> 7, s = t & 127;
    float ang = -2.0f * PI_F * (float)sfreq(r) * (float)s * (1.0f / 128.0f);
    wf_re[t] = cosf(ang) * (1.0f / 128.0f);
    wf_im[t] = sinf(ang) * (1.0f / 128.0f);
  } else {
    int u = t - 64 * 128;        // Wi: [s][r], 128x64
    int s = u >> 6, r = u & 63;
    float ang = 2.0f * PI_F * (float)sfreq(r) * (float)s * (1.0f / 128.0f);
    wi_re[u] = cosf(ang);
    wi_im[u] = sinf(ang);
  }
}

// ---- Stage A: forward DFT along s1, truncate to 64 modes -------------------
// T1[b][r1][s2][c] = sum_s1 Wf[r1][s1] * X[b][s1][s2][c]   (X real)
__global__ __launch_bounds__(256) void fwd_axis1(const float* __restrict__ X,
                                                 const float* __restrict__ wf_re, const float* __restrict__ wf_im,
                                                 float* __restrict__ t1_re, float* __restrict__ t1_im) {
  __shared__ __align__(16) float lX[128 * 128];   // [s1][c]  64KB
  __shared__ __align__(16) float lWr[64 * 128];   // 32KB
  __shared__ __align__(16) float lWi[64 * 128];   // 32KB
  int b = blockIdx.x >> 7;
  int s2 = blockIdx.x & 127;
  int tid = threadIdx.x;
  {
    const size_t base = (size_t)b * 2097152 + (size_t)s2 * 128;   // floats
    float4* dst = reinterpret_cast<float4*>(lX);
    for (int i4 = tid; i4 < 4096; i4 += 256) {                    // 32 float4 per s1 row
      int s1 = i4 >> 5, c4 = i4 & 31;
      dst[i4] = *reinterpret_cast<const float4*>(X + base + (size_t)s1 * 16384 + c4 * 4);
    }
    const float4* wr = reinterpret_cast<const float4*>(wf_re);
    const float4* wi = reinterpret_cast<const float4*>(wf_im);
    float4* dr = reinterpret_cast<float4*>(lWr);
    float4* di = reinterpret_cast<float4*>(lWi);
    for (int i4 = tid; i4 < 2048; i4 += 256) { dr[i4] = wr[i4]; di[i4] = wi[i4]; }
  }
  __syncthreads();
  int wave = tid >> 5, lane = tid & 31;
  for (int task = wave; task < 64; task += 8) {
    int part = task & 1;            // 0: re, 1: im
    int mt = (task >> 1) & 3;       // r1 tile
    int ct = task >> 3;             // c tile 0..7
    const float* W = part ? lWi : lWr;
    v8f acc = {};
#pragma unroll
    for (int k0 = 0; k0 < 128; k0 += 4) {
      v2f a = load_a(W + mt * 16 * 128, 128, k0, lane);
      v2f bb = load_b(lX, 128, k0, ct * 16, lane);
      acc = wmma4(a, bb, acc);
    }
    float* out = part ? t1_im : t1_re;
    int n = lane & 15, mb = (lane >> 4) * 8;
#pragma unroll
    for (int v = 0; v < 8; v++) {
      int r1 = mt * 16 + mb + v;
      out[(((size_t)b * 64 + r1) * 128 + s2) * 128 + ct * 16 + n] = acc[v];
    }
  }
}

// ---- Stage B: forward DFT along s2 (complex x complex), truncate -----------
// T2[b][r1][r2][c] = sum_s2 Wf[r2][s2] * T1[b][r1][s2][c]
__global__ __launch_bounds__(256) void fwd_axis2(const float* __restrict__ t1_re, const float* __restrict__ t1_im,
                                                 const float* __restrict__ wf_re, const float* __restrict__ wf_im,
                                                 float* __restrict__ t2_re, float* __restrict__ t2_im) {
  __shared__ __align__(16) float lTr[128 * 64];   // [s2][c-half] 32KB
  __shared__ __align__(16) float lTi[128 * 64];
  __shared__ __align__(16) float lWr[64 * 128];
  __shared__ __align__(16) float lWi[64 * 128];
  int x = blockIdx.x;
  int ch = x & 1; x >>= 1;
  int r1 = x & 63, b = x >> 6;
  int c0 = ch * 64;
  int tid = threadIdx.x;
  {
    const size_t base = ((size_t)b * 64 + r1) * 16384 + c0;       // floats
    float4* dtr = reinterpret_cast<float4*>(lTr);
    float4* dti = reinterpret_cast<float4*>(lTi);
    for (int i4 = tid; i4 < 2048; i4 += 256) {                    // 16 float4 per s2 row
      int s2 = i4 >> 4, c4 = i4 & 15;
      size_t g = base + (size_t)s2 * 128 + c4 * 4;
      dtr[i4] = *reinterpret_cast<const float4*>(t1_re + g);
      dti[i4] = *reinterpret_cast<const float4*>(t1_im + g);
    }
    const float4* wr = reinterpret_cast<const float4*>(wf_re);
    const float4* wi = reinterpret_cast<const float4*>(wf_im);
    float4* dr = reinterpret_cast<float4*>(lWr);
    float4* di = reinterpret_cast<float4*>(lWi);
    for (int i4 = tid; i4 < 2048; i4 += 256) { dr[i4] = wr[i4]; di[i4] = wi[i4]; }
  }
  __syncthreads();
  int wave = tid >> 5, lane = tid & 31;
  for (int task = wave; task < 16; task += 8) {
    int rt = task & 3, ct = task >> 2;       // r2 tile, c tile (within half)
    v8f arr = {}, ari = {}, air = {}, aii = {};
#pragma unroll
    for (int k0 = 0; k0 < 128; k0 += 4) {
      v2f awr = load_a(lWr + rt * 16 * 128, 128, k0, lane);
      v2f awi = load_a(lWi + rt * 16 * 128, 128, k0, lane);
      v2f btr = load_b(lTr, 64, k0, ct * 16, lane);
      v2f bti = load_b(lTi, 64, k0, ct * 16, lane);
      arr = wmma4(awr, btr, arr);
      aii = wmma4(awi, bti, aii);
      ari = wmma4(awr, bti, ari);
      air = wmma4(awi, btr, air);
    }
    int n = lane & 15, mb = (lane >> 4) * 8;
#pragma unroll
    for (int v = 0; v < 8; v++) {
      int r2 = rt * 16 + mb + v;
      size_t g = (((size_t)b * 64 + r1) * 64 + r2) * 128 + c0 + ct * 16 + n;
      t2_re[g] = arr[v] - aii[v];
      t2_im[g] = ari[v] + air[v];
    }
  }
}

// ---- Stage C: per-mode complex channel GEMM --------------------------------
// Ym[b,i] = sum_j K[e][i][j] * Xm[b,j] ; A = [Xre(8);Xim(8)] 16x128, B = K^T tiles
__global__ __launch_bounds__(256) void mode_gemm(const float* __restrict__ t2_re, const float* __restrict__ t2_im,
                                                 const float* __restrict__ Kb, // interleaved complex64 floats
                                                 float* __restrict__ t3_re, float* __restrict__ t3_im) {
  __shared__ __align__(16) float lA[16 * 128];        // rows 0-7: Xre, 8-15: Xim  (8KB)
  __shared__ __align__(16) float lKr[8 * 16 * 128];   // per-wave K^T tile [k=j][n=i], ld=16 (64KB)
  __shared__ __align__(16) float lKi[8 * 16 * 128];   // 64KB
  int m2 = blockIdx.x & 63, m1 = blockIdx.x >> 6;
  int fx = sfreq(m1), fy = sfreq(m2);
  int e = fx * fx + fy * fy;            // radius^2 index into kernel_base
  int tid = threadIdx.x, wave = tid >> 5, lane = tid & 31;
  {
    float4* dst = reinterpret_cast<float4*>(lA);
    for (int i4 = tid; i4 < 512; i4 += 256) {     // 32 float4 per A row
      int m = i4 >> 5, j4 = i4 & 31;
      size_t base = (((size_t)(m & 7) * 64 + m1) * 64 + m2) * 128;
      const float* src = (m < 8) ? t2_re : t2_im;
      dst[i4] = *reinterpret_cast<const float4*>(src + base + j4 * 4);
    }
  }
  {
    // per-wave K tile: i rows [wave*16, wave*16+16); read float4 = 2 complex, scatter transposed
    const float* Ke = Kb + (size_t)e * (128 * 128 * 2);
    float* kr = lKr + wave * 2048;
    float* ki = lKi + wave * 2048;
    for (int idx = lane; idx < 1024; idx += 32) {
      int il = idx >> 6;                 // i-local 0..15
      int j2 = (idx & 63) << 1;          // j base (even), covers j2, j2+1
      int i = wave * 16 + il;
      float4 q = *reinterpret_cast<const float4*>(Ke + ((size_t)i * 128 + j2) * 2);
      kr[j2 * 16 + il] = q.x;            // re(j2)
      ki[j2 * 16 + il] = q.y;            // im(j2)
      kr[(j2 + 1) * 16 + il] = q.z;      // re(j2+1)
      ki[(j2 + 1) * 16 + il] = q.w;      // im(j2+1)
    }
  }
  __syncthreads();
  v8f d1 = {}, d2 = {};
  const float* kr = lKr + wave * 2048;
  const float* ki = lKi + wave * 2048;
#pragma unroll
  for (int k0 = 0; k0 < 128; k0 += 4) {
    v2f a = load_a(lA, 128, k0, lane);
    v2f br = load_b(kr, 16, k0, 0, lane);
    v2f bi = load_b(ki, 16, k0, 0, lane);
    d1 = wmma4(a, br, d1);
    d2 = wmma4(a, bi, d2);
  }
  int n = lane & 15, i0 = wave * 16;
#pragma unroll
  for (int v = 0; v < 8; v++) {
    float d2sw = __shfl_xor(d2[v], 16, 32);   // row v+8 partner (Xim*Kim)
    float d1sw = __shfl_xor(d1[v], 16, 32);   // row v+8 partner (Xim*Kre)
    if (lane < 16) {                          // lanes 0-15 hold rows m=v (b=v)
      size_t g = (((size_t)v * 64 + m1) * 64 + m2) * 128 + i0 + n;
      t3_re[g] = d1[v] - d2sw;                // XreKre - XimKim
      t3_im[g] = d1sw + d2[v];                // XimKre + XreKim
    }
  }
}

// ---- Stage D: inverse DFT along s2 (zero-pad folded into Wi 128x64) --------
// T4[b][r1][s2][c] = sum_r2 Wi[s2][r2] * T3[b][r1][r2][c]
__global__ __launch_bounds__(256) void inv_axis2(const float* __restrict__ t3_re, const float* __restrict__ t3_im,
                                                 const float* __restrict__ wi_re, const float* __restrict__ wi_im,
                                                 float* __restrict__ t4_re, float* __restrict__ t4_im) {
  __shared__ __align__(16) float lTr[64 * 128];   // [r2][c] 32KB
  __shared__ __align__(16) float lTi[64 * 128];
  __shared__ __align__(16) float lIr[128 * 64];   // [s2][r2] 32KB
  __shared__ __align__(16) float lIi[128 * 64];
  int r1 = blockIdx.x & 63, b = blockIdx.x >> 6;
  int tid = threadIdx.x;
  {
    const size_t base = ((size_t)b * 64 + r1) * 8192;             // floats
    float4* dtr = reinterpret_cast<float4*>(lTr);
    float4* dti = reinterpret_cast<float4*>(lTi);
    float4* dir = reinterpret_cast<float4*>(lIr);
    float4* dii = reinterpret_cast<float4*>(lIi);
    const float4* sir = reinterpret_cast<const float4*>(wi_re);
    const float4* sii = reinterpret_cast<const float4*>(wi_im);
    for (int i4 = tid; i4 < 2048; i4 += 256) {
      dtr[i4] = *reinterpret_cast<const float4*>(t3_re + base + i4 * 4);
      dti[i4] = *reinterpret_cast<const float4*>(t3_im + base + i4 * 4);
      dir[i4] = sir[i4];
      dii[i4] = sii[i4];
    }
  }
  __syncthreads();
  int wave = tid >> 5, lane = tid & 31;
  for (int task = wave; task < 64; task += 8) {
    int st = task & 7, ct = task >> 3;        // s2 tile, c tile
    v8f arr = {}, ari = {}, air = {}, aii = {};
#pragma unroll
    for (int k0 = 0; k0 < 64; k0 += 4) {
      v2f ar = load_a(lIr + st * 16 * 64, 64, k0, lane);
      v2f ai = load_a(lIi + st * 16 * 64, 64, k0, lane);
      v2f br = load_b(lTr, 128, k0, ct * 16, lane);
      v2f bi = load_b(lTi, 128, k0, ct * 16, lane);
      arr = wmma4(ar, br, arr);
      aii = wmma4(ai, bi, aii);
      ari = wmma4(ar, bi, ari);
      air = wmma4(ai, br, air);
    }
    int n = lane & 15, mb = (lane >> 4) * 8;
#pragma unroll
    for (int v = 0; v < 8; v++) {
      int s2 = st * 16 + mb + v;
      size_t g = (((size_t)b * 64 + r1) * 128 + s2) * 128 + ct * 16 + n;
      t4_re[g] = arr[v] - aii[v];
      t4_im[g] = ari[v] + air[v];
    }
  }
}

// ---- Stage E: inverse DFT along s1, real part only -------------------------
// out[b][s1][s2][c] = Re( sum_r1 Wi[s1][r1] * T4[b][r1][s2][c] )
__global__ __launch_bounds__(256) void inv_axis1_real(const float* __restrict__ t4_re, const float* __restrict__ t4_im,
                                                      const float* __restrict__ wi_re, const float* __restrict__ wi_im,
                                                      float* __restrict__ out) {
  __shared__ __align__(16) float lTr[64 * 128];   // [r1][c]
  __shared__ __align__(16) float lTi[64 * 128];
  __shared__ __align__(16) float lIr[128 * 64];   // [s1][r1]
  __shared__ __align__(16) float lIi[128 * 64];
  int s2 = blockIdx.x & 127, b = blockIdx.x >> 7;
  int tid = threadIdx.x;
  {
    const size_t base = (size_t)b * 1048576 + (size_t)s2 * 128;   // floats; r1 stride 16384
    float4* dtr = reinterpret_cast<float4*>(lTr);
    float4* dti = reinterpret_cast<float4*>(lTi);
    float4* dir = reinterpret_cast<float4*>(lIr);
    float4* dii = reinterpret_cast<float4*>(lIi);
    const float4* sir = reinterpret_cast<const float4*>(wi_re);
    const float4* sii = reinterpret_cast<const float4*>(wi_im);
    for (int i4 = tid; i4 < 2048; i4 += 256) {
      int r1 = i4 >> 5, c4 = i4 & 31;
      size_t g = base + (size_t)r1 * 16384 + c4 * 4;
      dtr[i4] = *reinterpret_cast<const float4*>(t4_re + g);
      dti[i4] = *reinterpret_cast<const float4*>(t4_im + g);
      dir[i4] = sir[i4];
      dii[i4] = sii[i4];
    }
  }
  __syncthreads();
  int wave = tid >> 5, lane = tid & 31;
  for (int task = wave; task < 64; task += 8) {
    int st = task & 7, ct = task >> 3;        // s1 tile, c tile
    v8f accr = {}, acci = {};
#pragma unroll
    for (int k0 = 0; k0 < 64; k0 += 4) {
      v2f ar = load_a(lIr + st * 16 * 64, 64, k0, lane);
      v2f ai = load_a(lIi + st * 16 * 64, 64, k0, lane);
      v2f br = load_b(lTr, 128, k0, ct * 16, lane);
      v2f bi = load_b(lTi, 128, k0, ct * 16, lane);
      accr = wmma4(ar, br, accr);
      acci = wmma4(ai, bi, acci);
    }
    int n = lane & 15, mb = (lane >> 4) * 8;
#pragma unroll
    for (int v = 0; v < 8; v++) {
      int s1 = st * 16 + mb + v;
      out[(((size_t)b * 128 + s1) * 128 + s2) * 128 + ct * 16 + n] = accr[v] - acci[v];
    }
  }
}

extern "C" void kernel_launch(void* const* d_in, const int* in_sizes, int n_in,
                              void* d_out, int out_size, void* d_ws, size_t ws_size,
                              hipStream_t stream) {
  (void)in_sizes; (void)n_in; (void)out_size; (void)ws_size;
  const float* X  = (const float*)d_in[0];   // (8,128,128,128) f32
  const float* Kb = (const float*)d_in[1];   // (2049,128,128) complex64 interleaved
  float* out = (float*)d_out;

  float* ws = (float*)d_ws;
  float* wf_re = ws;                               // 64*128
  float* wf_im = wf_re + 64 * 128;
  float* wi_re = wf_im + 64 * 128;                 // 128*64
  float* wi_im = wi_re + 128 * 64;
  const size_t n1 = (size_t)8 * 64 * 128 * 128;    // 8.39M floats
  const size_t n2 = (size_t)8 * 64 * 64 * 128;     // 4.19M floats
  float* t1_re = wi_im + 128 * 64;
  float* t1_im = t1_re + n1;
  float* t2_re = t1_im + n1;
  float* t2_im = t2_re + n2;
  float* t3_re = t2_im + n2;
  float* t3_im = t3_re + n2;
  float* t4_re = t1_re;   // alias: t1 dead after stage B
  float* t4_im = t1_im;

  hipLaunchKernelGGL(init_twiddles, dim3(64), dim3(256), 0, stream, wf_re, wf_im, wi_re, wi_im);
  hipLaunchKernelGGL(fwd_axis1,     dim3(8 * 128),    dim3(256), 0, stream, X, wf_re, wf_im, t1_re, t1_im);
  hipLaunchKernelGGL(fwd_axis2,     dim3(8 * 64 * 2), dim3(256), 0, stream, t1_re, t1_im, wf_re, wf_im, t2_re, t2_im);
  hipLaunchKernelGGL(mode_gemm,     dim3(64 * 64),    dim3(256), 0, stream, t2_re, t2_im, Kb, t3_re, t3_im);
  hipLaunchKernelGGL(inv_axis2,     dim3(8 * 64),     dim3(256), 0, stream, t3_re, t3_im, wi_re, wi_im, t4_re, t4_im);
  hipLaunchKernelGGL(inv_axis1_real,dim3(8 * 128),    dim3(256), 0, stream, t4_re, t4_im, wi_re, wi_im, out);
}